// Operator_Layer_Grouped_81295140978842
// MI455X (gfx1250) — compile-verified
//
#include <hip/hip_runtime.h>

// CDNA5 / gfx1250, wave32. Grouped poly-conv via V_WMMA_F32_16X16X4_F32.
// Problem constants from the reference.
#define C_IN   128
#define HW     128
#define QPOW   3
#define OUT_CH 512
#define HOUT   126   // 128 - 3 + 1 (VALID 3x3)

typedef __attribute__((ext_vector_type(2))) float v2f;
typedef __attribute__((ext_vector_type(8))) float v8f;

// Per wave: one (b, g, 16-pixel output chunk).
// D(16x16) = sum_{s=0..6} A_s(16x4) * B_s(4x16) + C(bias)
//   A rows 0..3  = weights of the 4 output channels of group g over taps
//   B cols 0..15 = powered inputs at taps for 16 consecutive output pixels
//   taps t = r*9 + ky*3 + kx, t=0..26, K padded to 28 (tap 27 = 0)
__global__ __launch_bounds__(128)
void polyconv_wmma_kernel(const float* __restrict__ x,
                          const float* __restrict__ w,
                          const float* __restrict__ bias,
                          float* __restrict__ out)
{
    const int lane  = threadIdx.x & 31;
    const int wave  = threadIdx.x >> 5;
    const int chunk = blockIdx.x * 4 + wave;   // 0..1007
    const int y     = chunk >> 3;              // output row 0..125
    const int x0    = (chunk & 7) << 4;        // output col base 0,16,...,112
    const int g     = blockIdx.y;              // group 0..127
    const int b     = blockIdx.z;              // batch 0..15

    // Resolve the powers-permutation for the 3 input channels of group g:
    // permuted channel i=3g+r -> powers channel j=(i>>7)+3*(i&127)
    //   -> original channel j&127 at power (j>>7)+1
    const float* base0; const float* base1; const float* base2;
    int pw0, pw1, pw2;
    {
        int i0 = 3 * g + 0, i1 = 3 * g + 1, i2 = 3 * g + 2;
        int j0 = (i0 >> 7) + 3 * (i0 & 127);
        int j1 = (i1 >> 7) + 3 * (i1 & 127);
        int j2 = (i2 >> 7) + 3 * (i2 & 127);
        pw0 = (j0 >> 7) + 1; pw1 = (j1 >> 7) + 1; pw2 = (j2 >> 7) + 1;
        base0 = x + (size_t)(b * C_IN + (j0 & 127)) * (HW * HW);
        base1 = x + (size_t)(b * C_IN + (j1 & 127)) * (HW * HW);
        base2 = x + (size_t)(b * C_IN + (j2 & 127)) * (HW * HW);
    }

    const int n     = lane & 15;   // pixel column within chunk; also A row m
    const int khalf = lane >> 4;   // K half-select (ISA f32 16x4 layout)

    // C/D layout: VGPR m, lanes 0..15 hold M=m, N=lane. Seed rows 0..3 with bias.
    v8f acc;
#pragma unroll
    for (int m = 0; m < 8; ++m) acc[m] = 0.0f;
    if (lane < 16) {
#pragma unroll
        for (int m = 0; m < 4; ++m) acc[m] = bias[4 * g + m];
    }

    // 7 chained K=4 WMMA steps covering taps 0..27 (tap 27 zero-padded).
#pragma unroll
    for (int s = 0; s < 7; ++s) {
        const int t0 = 4 * s + 2 * khalf;  // A/B VGPR0 tap; VGPR1 tap = t0+1
        v2f av, bv;
#pragma unroll
        for (int jj = 0; jj < 2; ++jj) {
            const int t = t0 + jj;
            float a = 0.0f, bval = 0.0f;
            if (t < 27) {
                // A: weight[(4g+m), t], rows m>=4 are zero
                if (n < 4) a = w[(4 * g + n) * 27 + t];
                // B: powered input at tap t, pixel x0+n
                const int r  = t / 9;
                const int k  = t - 9 * r;
                const int ky = k / 3;
                const int kx = k - 3 * ky;
                int xi = x0 + n + kx;
                if (xi > HW - 1) xi = HW - 1;   // clamp: feeds only unused lanes
                const float* bp = base0; int p = pw0;
                if (r == 1) { bp = base1; p = pw1; }
                else if (r == 2) { bp = base2; p = pw2; }
                float v  = bp[(y + ky) * HW + xi];
                float vv = v;
                if (p >= 2) vv *= v;
                if (p == 3) vv *= v;
                bval = vv;
            }
            av[jj] = a;
            bv[jj] = bval;
        }
        // v_wmma_f32_16x16x4_f32: D = A*B + C  (8-arg form)
        acc = __builtin_amdgcn_wmma_f32_16x16x4_f32(
            /*neg_a=*/false, av, /*neg_b=*/false, bv,
            /*c_mod=*/(short)0, acc, /*reuse_a=*/false, /*reuse_b=*/false);
    }

    // Store rows 0..3 (the 4 real output channels), lanes 0..15, edge-masked.
    const int xo = x0 + n;
    if (lane < 16 && xo < HOUT) {
#pragma unroll
        for (int m = 0; m < 4; ++m) {
            out[((size_t)(b * OUT_CH + 4 * g + m) * HOUT + y) * HOUT + xo] = acc[m];
        }
    }
}

extern "C" void kernel_launch(void* const* d_in, const int* in_sizes, int n_in,
                              void* d_out, int out_size, void* d_ws, size_t ws_size,
                              hipStream_t stream)
{
    (void)in_sizes; (void)n_in; (void)out_size; (void)d_ws; (void)ws_size;
    const float* x    = (const float*)d_in[0];
    const float* w    = (const float*)d_in[1];
    const float* bias = (const float*)d_in[2];
    float* out = (float*)d_out;

    // 1008 chunks (126 rows x 8 col-chunks) / 4 waves per block = 252 blocks in x
    dim3 grid(252, 128, 16);
    polyconv_wmma_kernel<<<grid, dim3(128, 1, 1), 0, stream>>>(x, w, bias, out);
}